// LSTM_LSTM_51247549776562
// MI455X (gfx1250) — compile-verified
//
#include <hip/hip_runtime.h>
#include <math.h>

typedef float v2f __attribute__((ext_vector_type(2)));
typedef float v8f __attribute__((ext_vector_type(8)));

#define WMMA_F32X4(a, b, c) \
    __builtin_amdgcn_wmma_f32_16x16x4_f32(false, (a), false, (b), (short)0, (c), false, false)

// branch-free sigmoid / tanh built on v_exp_f32 / v_rcp_f32
__device__ __forceinline__ float sigf(float x)
{
    return __builtin_amdgcn_rcpf(1.0f + __expf(-x));
}
__device__ __forceinline__ float tanh_fast(float x)
{
    // 1 - 2/(e^{2x}+1); saturates correctly for |x| large (inf -> 1, 0 -> -1)
    return 1.0f - 2.0f * __builtin_amdgcn_rcpf(__expf(2.0f * x) + 1.0f);
}

// ---------------------------------------------------------------------------
// K1: encoder input projection  gx_e[t, g] = x[t,:8] @ Wih_e[g,:8] + (bih+bhh)[g]
//     [T,8] @ [8,12] -> [T,16].  Cols 12..15 are don't-care (never read).
// ---------------------------------------------------------------------------
__global__ __launch_bounds__(32)
void k_enc_gates(const float* __restrict__ x, const float* __restrict__ Wih_e,
                 const float* __restrict__ bih_e, const float* __restrict__ bhh_e,
                 float* __restrict__ gx_e, int T)
{
    const int lane  = threadIdx.x;
    const int row0  = blockIdx.x * 16;
    const int mrow  = lane & 15;
    const int khalf = (lane >> 4) * 2;
    const int col   = lane & 15;
    const int rhi   = 8 * (lane >> 4);

    int ar = row0 + mrow; if (ar > T - 1) ar = T - 1;
    const int cc = (col < 12) ? col : 0;      // clamp; cols 12..15 produce garbage (unread)

    v8f c;
    const float bias = bih_e[cc] + bhh_e[cc];
#pragma unroll
    for (int r = 0; r < 8; ++r) c[r] = bias;

#pragma unroll
    for (int kc = 0; kc < 2; ++kc) {          // K = 8
        const int kb = kc * 4 + khalf;
        v2f a, b;
        a[0] = x[ar * 8 + kb + 0];
        a[1] = x[ar * 8 + kb + 1];
        b[0] = Wih_e[cc * 8 + kb + 0];
        b[1] = Wih_e[cc * 8 + kb + 1];
        c = WMMA_F32X4(a, b, c);
    }

    if (row0 + 16 <= T) {
#pragma unroll
        for (int r = 0; r < 8; ++r) gx_e[(size_t)(row0 + r + rhi) * 16 + col] = c[r];
    } else {
#pragma unroll
        for (int r = 0; r < 8; ++r) {
            int row = row0 + r + rhi;
            if (row < T) gx_e[(size_t)row * 16 + col] = c[r];
        }
    }
}

// ---------------------------------------------------------------------------
// K2: decoder input projection (z-independent):
//     gx_d[t, g] = s[t,:6] @ Wih_d[g,:6] + (bih+bhh)[g] ; [T,6]@[6,40] -> [T,48]
//     Cols 40..47 are don't-care.  Only A needs true zero K-padding.
// ---------------------------------------------------------------------------
__global__ __launch_bounds__(32)
void k_dec_gates(const float* __restrict__ s, const float* __restrict__ Wih_d,
                 const float* __restrict__ bih_d, const float* __restrict__ bhh_d,
                 float* __restrict__ gx_d, int T)
{
    const int lane  = threadIdx.x;
    const int row0  = blockIdx.x * 16;
    const int mrow  = lane & 15;
    const int khalf = (lane >> 4) * 2;
    const int col16 = lane & 15;
    const int rhi   = 8 * (lane >> 4);

    int ar = row0 + mrow; if (ar > T - 1) ar = T - 1;

    v2f a[2];                                  // K = 6 padded to 8 (A padding must be 0)
#pragma unroll
    for (int kc = 0; kc < 2; ++kc) {
        const int kb = kc * 4 + khalf;
        const int k0 = (kb + 0 < 6) ? kb + 0 : 5;
        const int k1 = (kb + 1 < 6) ? kb + 1 : 5;
        float v0 = s[ar * 6 + k0];
        float v1 = s[ar * 6 + k1];
        a[kc][0] = (kb + 0 < 6) ? v0 : 0.f;
        a[kc][1] = (kb + 1 < 6) ? v1 : 0.f;
    }

#pragma unroll
    for (int nt = 0; nt < 3; ++nt) {
        const int n  = nt * 16 + col16;
        const int nn = (n < 40) ? n : 0;       // clamp; garbage in padded cols is fine
        v8f c;
        const float bias = bih_d[nn] + bhh_d[nn];
#pragma unroll
        for (int r = 0; r < 8; ++r) c[r] = bias;
#pragma unroll
        for (int kc = 0; kc < 2; ++kc) {
            const int kb = kc * 4 + khalf;
            const int k0 = (kb + 0 < 6) ? kb + 0 : 5;   // garbage ok: A is 0 there
            const int k1 = (kb + 1 < 6) ? kb + 1 : 5;
            v2f b;
            b[0] = Wih_d[nn * 7 + k0];
            b[1] = Wih_d[nn * 7 + k1];
            c = WMMA_F32X4(a[kc], b, c);
        }
        if (row0 + 16 <= T) {
#pragma unroll
            for (int r = 0; r < 8; ++r)
                gx_d[(size_t)(row0 + r + rhi) * 48 + nt * 16 + col16] = c[r];
        } else {
#pragma unroll
            for (int r = 0; r < 8; ++r) {
                int row = row0 + r + rhi;
                if (row < T) gx_d[(size_t)row * 48 + nt * 16 + col16] = c[r];
            }
        }
    }
}

// ---------------------------------------------------------------------------
// K3: encoder LSTM scan (H=3, 12 gates) + FC stack -> z.  One wave, T steps.
// Branch-free loop body; gate stream prefetched ~16 steps ahead into WGP$.
// ---------------------------------------------------------------------------
__global__ __launch_bounds__(32)
void k_enc_scan(const float* __restrict__ gx_e, const float* __restrict__ Whh_e,
                const float* __restrict__ W1e, const float* __restrict__ b1e,
                const float* __restrict__ W2e, const float* __restrict__ b2e,
                float* __restrict__ zbuf, int T)
{
    const int lane = threadIdx.x;
    const int gl   = (lane < 12) ? lane : 0;           // clamped gate index

    const float w0 = Whh_e[gl * 3 + 0];
    const float w1 = Whh_e[gl * 3 + 1];
    const float w2 = Whh_e[gl * 3 + 2];

    float h = 0.f, cst = 0.f;
    float gx = gx_e[gl];                               // step-0 gates (pipelined)

    for (int t = 0; t < T; ++t) {
        const int tn = (t + 1 < T) ? t + 1 : t;
        const float gx_next = gx_e[(size_t)tn * 16 + gl];      // issues early
        const int tp = (t + 16 < T) ? t + 16 : T - 1;
        __builtin_prefetch(&gx_e[(size_t)tp * 16 + gl], 0, 3); // global_prefetch

        const float h0 = __shfl(h, 0), h1 = __shfl(h, 1), h2 = __shfl(h, 2);
        const float gate = gx + w0 * h0 + w1 * h1 + w2 * h2;

        const float ig = gate;                       // lane j's i-gate is its own gate
        const float fg = __shfl(gate, lane + 3);
        const float gg = __shfl(gate, lane + 6);
        const float og = __shfl(gate, lane + 9);

        cst = sigf(fg) * cst + sigf(ig) * tanh_fast(gg);
        h   = sigf(og) * tanh_fast(cst);
        gx  = gx_next;
    }

    // encoder FC: h1e = relu(W1e @ h_last + b1e) [6] ; z = W2e @ h1e + b2e
    const float h0 = __shfl(h, 0), h1 = __shfl(h, 1), h2 = __shfl(h, 2);
    const int ml = (lane < 6) ? lane : 0;
    float h1e = fmaxf(0.f, W1e[ml*3+0]*h0 + W1e[ml*3+1]*h1 + W1e[ml*3+2]*h2 + b1e[ml]);
    float he[6];
#pragma unroll
    for (int m = 0; m < 6; ++m) he[m] = __shfl(h1e, m);        // all lanes active
    if (lane == 0) {
        float zv = b2e[0];
#pragma unroll
        for (int m = 0; m < 6; ++m) zv += W2e[m] * he[m];
        zbuf[0] = zv;
    }
}

// ---------------------------------------------------------------------------
// K4: decoder LSTM scan (H=10, 40 gates).  One wave; lane j<10 owns hidden j
// and its 4 gates.  Branch-free in-loop loads; deep prefetch; z folded in.
// Writes hs[t,0..9] (cols 10..15 zeroed for the WMMA MLP).
// ---------------------------------------------------------------------------
__global__ __launch_bounds__(32)
void k_dec_scan(const float* __restrict__ gx_d, const float* __restrict__ Whh_d,
                const float* __restrict__ Wih_d, const float* __restrict__ zbuf,
                float* __restrict__ hs, int T)
{
    const int lane = threadIdx.x;
    const int j    = (lane < 10) ? lane : 0;           // clamp; lanes >=10 shadow unit 0

    float wi[10], wf[10], wg[10], wo[10];
#pragma unroll
    for (int k = 0; k < 10; ++k) {
        wi[k] = Whh_d[(     j) * 10 + k];
        wf[k] = Whh_d[(10 + j) * 10 + k];
        wg[k] = Whh_d[(20 + j) * 10 + k];
        wo[k] = Whh_d[(30 + j) * 10 + k];
    }
    const float z   = zbuf[0];
    const float zbi = z * Wih_d[(     j) * 7 + 6];
    const float zbf = z * Wih_d[(10 + j) * 7 + 6];
    const float zbg = z * Wih_d[(20 + j) * 7 + 6];
    const float zbo = z * Wih_d[(30 + j) * 7 + 6];

    float h = 0.f, cst = 0.f;
    float hv[10];
#pragma unroll
    for (int k = 0; k < 10; ++k) hv[k] = 0.f;

    float pgi = gx_d[j], pgf = gx_d[10 + j], pgg = gx_d[20 + j], pgo = gx_d[30 + j];

    for (int t = 0; t < T; ++t) {
        float gi = pgi + zbi, gf = pgf + zbf, gg = pgg + zbg, go = pgo + zbo;

        // pipeline next step's gate loads (branch-free, clamped), + deep prefetch
        const int tn = (t + 1 < T) ? t + 1 : t;
        const float* gn = gx_d + (size_t)tn * 48;
        pgi = gn[j]; pgf = gn[10 + j]; pgg = gn[20 + j]; pgo = gn[30 + j];
        const int tp = (t + 16 < T) ? t + 16 : T - 1;
        __builtin_prefetch(gx_d + (size_t)tp * 48 + j, 0, 3);

#pragma unroll
        for (int k = 0; k < 10; ++k) {
            gi += wi[k] * hv[k]; gf += wf[k] * hv[k];
            gg += wg[k] * hv[k]; go += wo[k] * hv[k];
        }
        cst = sigf(gf) * cst + sigf(gi) * tanh_fast(gg);
        h   = sigf(go) * tanh_fast(cst);

        if (lane < 16) hs[(size_t)t * 16 + lane] = (lane < 10) ? h : 0.f;
#pragma unroll
        for (int k = 0; k < 10; ++k) hv[k] = __shfl(h, k);
    }
}

// ---------------------------------------------------------------------------
// K5: per-step MLP 10->20->20->2 over hs, WMMA f32, chained through LDS.
// LDS tile cols 20..31 are don't-care (written, never read as A inputs).
// ---------------------------------------------------------------------------
__global__ __launch_bounds__(32)
void k_mlp(const float* __restrict__ hs,
           const float* __restrict__ W1, const float* __restrict__ b1,
           const float* __restrict__ W2, const float* __restrict__ b2,
           const float* __restrict__ W3, const float* __restrict__ b3,
           float* __restrict__ out, int T)
{
    __shared__ float tile[16 * 32];
    const int lane  = threadIdx.x;
    const int row0  = blockIdx.x * 16;
    const int mrow  = lane & 15;
    const int khalf = (lane >> 4) * 2;
    const int col16 = lane & 15;
    const int rhi   = 8 * (lane >> 4);

    int ar = row0 + mrow; if (ar > T - 1) ar = T - 1;

    // ---- layer 1: A = hs tile (K=10; cols 10..15 of hs are true zeros)
    v2f a[3];
#pragma unroll
    for (int kc = 0; kc < 3; ++kc) {           // covers k = 0..11
        const int kb = kc * 4 + khalf;
        a[kc][0] = hs[(size_t)ar * 16 + kb + 0];
        a[kc][1] = hs[(size_t)ar * 16 + kb + 1];
    }
    const int n1 = (col16 < 4) ? (16 + col16) : col16;  // unused helper? no -- remove
    (void)n1;
#pragma unroll
    for (int nt = 0; nt < 2; ++nt) {
        const int n  = nt * 16 + col16;
        const int nn = (n < 20) ? n : 0;       // clamp; padded cols never read back
        v8f c;
        const float bias = b1[nn];
#pragma unroll
        for (int r = 0; r < 8; ++r) c[r] = bias;
#pragma unroll
        for (int kc = 0; kc < 3; ++kc) {
            const int kb = kc * 4 + khalf;
            const int k0 = (kb + 0 < 10) ? kb + 0 : 9;  // garbage ok: A is 0 at k>=10
            const int k1 = (kb + 1 < 10) ? kb + 1 : 9;
            v2f b;
            b[0] = W1[nn * 10 + k0];
            b[1] = W1[nn * 10 + k1];
            c = WMMA_F32X4(a[kc], b, c);
        }
#pragma unroll
        for (int r = 0; r < 8; ++r)
            tile[(r + rhi) * 32 + nt * 16 + col16] = fmaxf(c[r], 0.f);   // relu H1
    }
    __syncthreads();

    // ---- layer 2: A = H1 (16x20 of 16x32 LDS), K=20 -> 5 exact chunks
    v2f a2[5];
#pragma unroll
    for (int kc = 0; kc < 5; ++kc) {
        const int kb = kc * 4 + khalf;
        a2[kc][0] = tile[mrow * 32 + kb + 0];
        a2[kc][1] = tile[mrow * 32 + kb + 1];
    }
    __syncthreads();
#pragma unroll
    for (int nt = 0; nt < 2; ++nt) {
        const int n  = nt * 16 + col16;
        const int nn = (n < 20) ? n : 0;
        v8f c;
        const float bias = b2[nn];
#pragma unroll
        for (int r = 0; r < 8; ++r) c[r] = bias;
#pragma unroll
        for (int kc = 0; kc < 5; ++kc) {
            const int kb = kc * 4 + khalf;
            v2f b;
            b[0] = W2[nn * 20 + kb + 0];
            b[1] = W2[nn * 20 + kb + 1];
            c = WMMA_F32X4(a2[kc], b, c);
        }
#pragma unroll
        for (int r = 0; r < 8; ++r)
            tile[(r + rhi) * 32 + nt * 16 + col16] = fmaxf(c[r], 0.f);   // relu H2
    }
    __syncthreads();

    // ---- layer 3: A = H2 (16x20), B = W3^T (2 valid columns)
    v2f a3[5];
#pragma unroll
    for (int kc = 0; kc < 5; ++kc) {
        const int kb = kc * 4 + khalf;
        a3[kc][0] = tile[mrow * 32 + kb + 0];
        a3[kc][1] = tile[mrow * 32 + kb + 1];
    }
    {
        const int nn = (col16 < 2) ? col16 : 0;
        v8f c;
        const float bias = b3[nn];
#pragma unroll
        for (int r = 0; r < 8; ++r) c[r] = bias;
#pragma unroll
        for (int kc = 0; kc < 5; ++kc) {
            const int kb = kc * 4 + khalf;
            v2f b;
            b[0] = W3[nn * 20 + kb + 0];
            b[1] = W3[nn * 20 + kb + 1];
            c = WMMA_F32X4(a3[kc], b, c);
        }
        if (col16 < 2) {
            if (row0 + 16 <= T) {
#pragma unroll
                for (int r = 0; r < 8; ++r)
                    out[(size_t)(row0 + r + rhi) * 2 + col16] = c[r];
            } else {
#pragma unroll
                for (int r = 0; r < 8; ++r) {
                    int row = row0 + r + rhi;
                    if (row < T) out[(size_t)row * 2 + col16] = c[r];
                }
            }
        }
    }
}

// ---------------------------------------------------------------------------
extern "C" void kernel_launch(void* const* d_in, const int* in_sizes, int n_in,
                              void* d_out, int out_size, void* d_ws, size_t ws_size,
                              hipStream_t stream)
{
    const float* x     = (const float*)d_in[0];
    const float* s     = (const float*)d_in[1];
    const float* Wih_e = (const float*)d_in[2];
    const float* Whh_e = (const float*)d_in[3];
    const float* bih_e = (const float*)d_in[4];
    const float* bhh_e = (const float*)d_in[5];
    const float* W1e   = (const float*)d_in[6];
    const float* b1e   = (const float*)d_in[7];
    const float* W2e   = (const float*)d_in[8];
    const float* b2e   = (const float*)d_in[9];
    const float* Wih_d = (const float*)d_in[10];
    const float* Whh_d = (const float*)d_in[11];
    const float* bih_d = (const float*)d_in[12];
    const float* bhh_d = (const float*)d_in[13];
    const float* W1    = (const float*)d_in[14];
    const float* b1    = (const float*)d_in[15];
    const float* W2    = (const float*)d_in[16];
    const float* b2    = (const float*)d_in[17];
    const float* W3    = (const float*)d_in[18];
    const float* b3    = (const float*)d_in[19];

    const int T = in_sizes[0] / 8;            // x is [T,8]
    const int ntiles = (T + 15) / 16;

    float* ws   = (float*)d_ws;
    float* gx_e = ws;                          // [T,16]
    float* gx_d = gx_e + (size_t)T * 16;       // [T,48]
    float* zbuf = gx_d + (size_t)T * 48;       // [16]
    float* hs   = zbuf + 16;                   // [T,16]

    k_enc_gates<<<ntiles, 32, 0, stream>>>(x, Wih_e, bih_e, bhh_e, gx_e, T);
    k_dec_gates<<<ntiles, 32, 0, stream>>>(s, Wih_d, bih_d, bhh_d, gx_d, T);
    k_enc_scan <<<1,      32, 0, stream>>>(gx_e, Whh_e, W1e, b1e, W2e, b2e, zbuf, T);
    k_dec_scan <<<1,      32, 0, stream>>>(gx_d, Whh_d, Wih_d, zbuf, hs, T);
    k_mlp      <<<ntiles, 32, 0, stream>>>(hs, W1, b1, W2, b2, W3, b3, (float*)d_out, T);
}